// SPD_to_vec_5901285065061
// MI455X (gfx1250) — compile-verified
//
#include <hip/hip_runtime.h>

// ---------------------------------------------------------------------------
// SPD Frechet-mean + affine-invariant distance for CDNA5 (gfx1250).
//
// Two independent groups of 256 SPD 8x8 matrices per wave32, packed as the
// diagonal blocks of a 16x16 fp32 tile.  All matrix functions are matmul-only
// iterations mapped onto V_WMMA_F32_16X16X4_F32.
//
// Tiles are held natively in the WMMA *operand* layout ("O-layout"):
//   lane(col=lane&15, hi=lane>>4) holds o[2*kc+j] = elem(col, 4*kc+2*hi+j).
// Since nearly all tiles are symmetric, the same registers serve as both the
// A operand (rows) and B operand (columns) -> matmuls need NO LDS staging.
// Only the C-layout matmul output is converted back to O-layout via one
// wave-private LDS image (2x ds_store_b128 + 8x ds_load_b32), with optional
// fused symmetrization.
// ---------------------------------------------------------------------------

typedef __attribute__((ext_vector_type(2))) float v2f;
typedef __attribute__((ext_vector_type(4))) float v4f;
typedef __attribute__((ext_vector_type(8))) float v8f;

#define SEQ       256
#define NGROUPS   2048
#define NS_ITERS  9

struct WaveCtx {
  float* lds;   // 256-float wave-private staging image (column-major: [N*16+M])
  int col;      // lane & 15
  int hi;       // lane >> 4
  int lane;
};

// Per-8x8-block sum: block P lives in lanes {0-7,16-23}, Q in {8-15,24-31}.
__device__ inline float blk_sum(float x) {
  x += __shfl_xor(x, 1, 32);
  x += __shfl_xor(x, 2, 32);
  x += __shfl_xor(x, 4, 32);
  x += __shfl_xor(x, 16, 32);
  return x;
}

// Identity diag(I8,I8) in O-layout.
__device__ inline v8f ident_o(const WaveCtx& w) {
  v8f r = {};
#pragma unroll
  for (int kc = 0; kc < 4; ++kc)
#pragma unroll
    for (int j = 0; j < 2; ++j)
      r[2 * kc + j] = ((4 * kc + 2 * w.hi + j) == w.col) ? 1.0f : 0.0f;
  return r;
}

// Pure-register 16x16x16 matmul: A (rows) x B (cols), both in O-layout.
// 4 chained V_WMMA_F32_16X16X4_F32, result in C/D layout.
__device__ inline v8f wmma_oo(v8f A, v8f B) {
  v8f acc = {};
#pragma unroll
  for (int kc = 0; kc < 4; ++kc) {
    v2f a, b;
    a.x = A[2 * kc]; a.y = A[2 * kc + 1];
    b.x = B[2 * kc]; b.y = B[2 * kc + 1];
    // (neg_a, A, neg_b, B, c_mod, C, reuse_a, reuse_b)
    acc = __builtin_amdgcn_wmma_f32_16x16x4_f32(false, a, false, b,
                                                (short)0, acc, false, false);
  }
  return acc;
}

// Stage a C-layout tile into the column-major image img[N*16 + M].
// C-layout: lane holds column N=col, rows M = v + 8*hi -> 8 consecutive
// floats in the image -> two ds_store_b128.
__device__ inline void stage_c(const WaveCtx& w, v8f C) {
  union { v8f v; v4f q[2]; } u;
  u.v = C;
  v4f* dst = (v4f*)(w.lds + w.col * 16 + 8 * w.hi);
  dst[0] = u.q[0];
  dst[1] = u.q[1];
}

// Read staged image back in O-layout row-form: elem(col, k) = img[k*16+col].
__device__ inline v8f read_row(const WaveCtx& w) {
  v8f r;
#pragma unroll
  for (int kc = 0; kc < 4; ++kc) {
    const int k0 = 4 * kc + 2 * w.hi;
    r[2 * kc]     = w.lds[(k0 + 0) * 16 + w.col];
    r[2 * kc + 1] = w.lds[(k0 + 1) * 16 + w.col];
  }
  return r;
}

// Fused symmetrize + convert: 0.5*(elem(col,k) + elem(k,col)).
// Transposed entries elem(k,col) = img[col*16+k] are contiguous -> b64 loads.
__device__ inline v8f read_symm(const WaveCtx& w) {
  v8f r;
#pragma unroll
  for (int kc = 0; kc < 4; ++kc) {
    const int k0 = 4 * kc + 2 * w.hi;
    v2f t = *(const v2f*)(w.lds + w.col * 16 + k0);
    r[2 * kc]     = 0.5f * (w.lds[(k0 + 0) * 16 + w.col] + t.x);
    r[2 * kc + 1] = 0.5f * (w.lds[(k0 + 1) * 16 + w.col] + t.y);
  }
  return r;
}

__device__ inline v8f mm(const WaveCtx& w, v8f A, v8f B) {
  v8f C = wmma_oo(A, B);
  stage_c(w, C);
  return read_row(w);
}

__device__ inline v8f mm_symm(const WaveCtx& w, v8f A, v8f B) {
  v8f C = wmma_oo(A, B);
  stage_c(w, C);
  return read_symm(w);
}

// Coupled Newton-Schulz: Ys ~= A^{1/2}, Zs ~= A^{-1/2} (A SPD, O-layout).
__device__ inline void ns_sqrt(const WaveCtx& w, v8f A, v8f& Ys, v8f& Zs) {
  float loc = 0.0f;
#pragma unroll
  for (int v = 0; v < 8; ++v) loc += A[v] * A[v];
  float f = sqrtf(blk_sum(loc));           // per-block Frobenius norm
  f = fmaxf(f, 1e-20f);
  const float rc = 1.0f / f;

  v8f I = ident_o(w);
  v8f Y = A * rc;
  v8f Z = I;
#pragma unroll 1
  for (int it = 0; it < NS_ITERS; ++it) {
    v8f T = I * 1.5f - mm(w, Z, Y) * 0.5f;   // 0.5*(3I - Z*Y)   (symmetric)
    Y = mm(w, Y, T);
    Z = mm(w, T, Z);
  }
  const float sf = sqrtf(f);
  Ys = Y * sf;
  Zs = Z * (1.0f / sf);
}

// log(S), S SPD: R = S^{1/4} via two NS sqrts, E = R - I,
// log R = E(I - E(I/2 - E(I/3 - E/4))), log S = 4 log R.
__device__ inline v8f spd_log(const WaveCtx& w, v8f S) {
  v8f Y, Z;
  ns_sqrt(w, S, Y, Z);   // Y = S^{1/2}
  ns_sqrt(w, Y, Y, Z);   // Y = S^{1/4}
  v8f I = ident_o(w);
  v8f E = Y - I;
  v8f T = I * (1.0f / 3.0f) - E * 0.25f;
  T = I * 0.5f - mm(w, E, T);
  T = I - mm(w, E, T);
  v8f LR = mm(w, E, T);
  return LR * 4.0f;
}

// S^p = exp(p log S): 4-term Taylor of Q = (p/4) log S, squared twice.
__device__ inline v8f spd_pow(const WaveCtx& w, v8f S, float p) {
  v8f L = spd_log(w, S);
  v8f I = ident_o(w);
  v8f Q = L * (0.25f * p);
  v8f inner = I * 0.5f + Q * (1.0f / 6.0f);
  v8f T = I + mm(w, Q, inner);
  v8f R = I + mm(w, Q, T);
  R = mm(w, R, R);
  R = mm(w, R, R);
  return R;
}

// Load diag(P,Q) tile straight into O-layout: every lane does two 8-byte
// loads of its row-chunk; off-diagonal slots are exact zeros.
__device__ inline v8f load_tile_o(const float* __restrict__ x, long bP, long bQ,
                                  const WaveCtx& w) {
  const bool isP = (w.col < 8);
  const float* base = x + (isP ? bP : bQ) + (long)(w.col & 7) * 8;
  v2f a = *(const v2f*)(base + 2 * w.hi);       // elem(row, {2hi, 2hi+1})
  v2f b = *(const v2f*)(base + 4 + 2 * w.hi);   // elem(row, {4+2hi, 5+2hi})
  v8f r = {};
  if (isP) { r[0] = a.x; r[1] = a.y; r[2] = b.x; r[3] = b.y; }  // k < 8
  else     { r[4] = a.x; r[5] = a.y; r[6] = b.x; r[7] = b.y; }  // k >= 8
  return r;
}

__global__ __launch_bounds__(256) void spd_fm_dist_kernel(
    const float* __restrict__ x, float* __restrict__ out) {
  __shared__ float lds[8 * 256];   // one 1KB staging image per wave

  const int lane = threadIdx.x & 31;
  const int wav  = threadIdx.x >> 5;
  const int wid  = blockIdx.x * 8 + wav;          // 0..1023
  const long g0 = 2L * wid;
  const long g1 = 2L * wid + 1;

  WaveCtx w;
  w.lds  = &lds[wav * 256];
  w.lane = lane;
  w.col  = lane & 15;
  w.hi   = lane >> 4;

  const long baseP = g0 * (SEQ * 64L);
  const long baseQ = g1 * (SEQ * 64L);

  // ---- recursive Frechet mean scan: M_1 = X_1, M_k = M #_{1/k} X_k -------
  v8f M = load_tile_o(x, baseP, baseQ, w);

#pragma unroll 1
  for (int k = 1; k < SEQ; ++k) {
    if (k + 1 < SEQ) {
      __builtin_prefetch(x + baseP + (long)(k + 1) * 64, 0, 0);
      __builtin_prefetch(x + baseQ + (long)(k + 1) * 64, 0, 0);
    }
    v8f X = load_tile_o(x, baseP + (long)k * 64, baseQ + (long)k * 64, w);
    const float wk = 1.0f / (float)(k + 1);

    v8f Msq, Misq;
    ns_sqrt(w, M, Msq, Misq);                 // M^{1/2}, M^{-1/2}
    v8f T1 = mm(w, Misq, X);                  // non-symmetric, A-operand only
    v8f S  = mm_symm(w, T1, Misq);            // sym(M^{-1/2} X M^{-1/2})
    v8f Sw = spd_pow(w, S, wk);               // S^{1/(k+1)}
    v8f T2 = mm(w, Msq, Sw);
    M = mm_symm(w, T2, Msq);                  // sym(M^{1/2} S^w M^{1/2})
  }

  // ---- distances: trace(log(X^{-1} FM)^2) = ||log(X^{-1/2} FM X^{-1/2})||_F^2
#pragma unroll 1
  for (int s = 0; s < SEQ; ++s) {
    if (s + 1 < SEQ) {
      __builtin_prefetch(x + baseP + (long)(s + 1) * 64, 0, 0);
      __builtin_prefetch(x + baseQ + (long)(s + 1) * 64, 0, 0);
    }
    v8f X = load_tile_o(x, baseP + (long)s * 64, baseQ + (long)s * 64, w);
    v8f Y, Z;
    ns_sqrt(w, X, Y, Z);                      // Z = X^{-1/2}
    v8f T1 = mm(w, Z, M);
    v8f A  = mm_symm(w, T1, Z);
    v8f L  = spd_log(w, A);

    float loc = 0.0f;
#pragma unroll
    for (int v = 0; v < 8; ++v) loc += L[v] * L[v];   // O-layout covers each
    const float d = blk_sum(loc);                     // element exactly once

    if (lane == 0) out[g0 * SEQ + s] = d;             // block P (col==0)
    if (lane == 8) out[g1 * SEQ + s] = d;             // block Q (col==8)
  }
}

extern "C" void kernel_launch(void* const* d_in, const int* in_sizes, int n_in,
                              void* d_out, int out_size, void* d_ws, size_t ws_size,
                              hipStream_t stream) {
  (void)in_sizes; (void)n_in; (void)d_ws; (void)ws_size; (void)out_size;
  const float* x = (const float*)d_in[0];
  float* out = (float*)d_out;
  // 2048 groups / 2 per wave = 1024 waves; 8 waves (256 threads) per block.
  dim3 grid(NGROUPS / 16);
  dim3 block(256);
  hipLaunchKernelGGL(spd_fm_dist_kernel, grid, block, 0, stream, x, out);
}